// SimpleRNN_43774306681487
// MI455X (gfx1250) — compile-verified
//
#include <hip/hip_runtime.h>
#include <hip/hip_bf16.h>
#include <math.h>

typedef float v2f __attribute__((ext_vector_type(2)));
typedef float v8f __attribute__((ext_vector_type(8)));

#define TT   128
#define BB   64
#define II   2048
#define HH   512
#define G4   2048          // 4*HH
#define NC   11

// ---------------------------------------------------------------------------
// CDNA5 async memory->LDS staging (ASYNCcnt-tracked, no VGPR landing zone).
// ---------------------------------------------------------------------------
#if defined(__has_builtin)
#  if __has_builtin(__builtin_amdgcn_s_wait_asynccnt)
#    define WAITA(n) __builtin_amdgcn_s_wait_asynccnt(n)
#  endif
#endif
#ifndef WAITA
#  define WAITA(n) asm volatile("s_wait_asynccnt %0" :: "n"(n) : "memory")
#endif

__device__ __forceinline__ unsigned lds_off32(const void* p) {
  // Generic pointers to LDS carry the LDS offset in the low 32 bits (aperture map).
  return (unsigned)(unsigned long long)p;
}

__device__ __forceinline__ void async_b128(unsigned lds, const float* g) {
  asm volatile("global_load_async_to_lds_b128 %0, %1, off"
               :: "v"(lds), "v"((unsigned long long)g)
               : "memory");
}

// ---------------------------------------------------------------------------
// Kernel 1: xg[m, n] = x[m, :] . W_ih[n, :] + b_ih[n] + b_hh[n]
//   M = T*B = 8192, N = 4H = 2048, K = I = 2048, fp32 via V_WMMA_F32_16X16X4_F32
//   WG 256 thr (8 wave32), tile 64(M) x 64(N), K-chunk 32, double-buffered
//   async-to-LDS staging; last chunk peeled to keep the hot loop branch-free.
// ---------------------------------------------------------------------------
__global__ __launch_bounds__(256) void xgates_gemm(
    const float* __restrict__ x, const float* __restrict__ W_ih,
    const float* __restrict__ b_ih, const float* __restrict__ b_hh,
    float* __restrict__ xg)
{
  __shared__ float As[2][64][36];   // x tile    [m][k], stride 36 (16B aligned, conflict-free)
  __shared__ float Bs[2][64][36];   // W_ih tile [n][k]  (== B^T, contiguous in k globally)

  const int tid  = threadIdx.x;
  const int wv   = tid >> 5;
  const int lane = tid & 31;
  const int lrow = lane & 15;
  const int lhi  = lane >> 4;

  const int n0 = blockIdx.x * 64;
  const int m0 = blockIdx.y * 64;

  const int m_sub = (wv & 3) * 16;   // 4 M sub-tiles
  const int n_sub = (wv >> 2) * 32;  // 2 N groups, 2 tiles each

  // per-thread staging slots: 2 float4 for A, 2 for B per chunk
  const int r0 = tid >> 3,          kq0 = (tid & 7) * 4;          // rows 0..31
  const int r1 = (tid + 256) >> 3,  kq1 = ((tid + 256) & 7) * 4;  // rows 32..63

  auto stage = [&](int buf, int k0) {
    async_b128(lds_off32(&As[buf][r0][kq0]), x    + (size_t)(m0 + r0) * II + k0 + kq0);
    async_b128(lds_off32(&As[buf][r1][kq1]), x    + (size_t)(m0 + r1) * II + k0 + kq1);
    async_b128(lds_off32(&Bs[buf][r0][kq0]), W_ih + (size_t)(n0 + r0) * II + k0 + kq0);
    async_b128(lds_off32(&Bs[buf][r1][kq1]), W_ih + (size_t)(n0 + r1) * II + k0 + kq1);
  };

  v8f acc0 = {}; v8f acc1 = {};

  auto compute = [&](int buf) {
#pragma unroll
    for (int ks = 0; ks < 8; ++ks) {
      const int kk = ks * 4 + lhi * 2;            // per-lane K base (ISA 16x4 fp32 A layout)
      const v2f a  = *(const v2f*)&As[buf][m_sub + lrow][kk];
      const v2f b0 = *(const v2f*)&Bs[buf][n_sub      + lrow][kk];
      const v2f b1 = *(const v2f*)&Bs[buf][n_sub + 16 + lrow][kk];
      acc0 = __builtin_amdgcn_wmma_f32_16x16x4_f32(false, a, false, b0, (short)0, acc0, false, false);
      acc1 = __builtin_amdgcn_wmma_f32_16x16x4_f32(false, a, false, b1, (short)0, acc1, false, false);
    }
  };

  stage(0, 0);
  // steady state: branch-free, always prefetch the next chunk
  for (int k0 = 0; k0 < II - 32; k0 += 32) {
    const int buf = (k0 >> 5) & 1;
    stage(buf ^ 1, k0 + 32);
    WAITA(4);                 // 4 outstanding = next buffer; current buffer resident
    __syncthreads();
    compute(buf);
    __syncthreads();          // all waves done reading buf before it is overwritten
  }
  // peeled final chunk
  WAITA(0);
  __syncthreads();
  compute(((II - 32) >> 5) & 1);

  const int c0 = n0 + n_sub + lrow;
  const int c1 = c0 + 16;
  const float bias0 = b_ih[c0] + b_hh[c0];
  const float bias1 = b_ih[c1] + b_hh[c1];
#pragma unroll
  for (int r = 0; r < 8; ++r) {                   // C layout: VGPR r -> M = r + lhi*8
    const int grow = m0 + m_sub + r + lhi * 8;
    xg[(size_t)grow * G4 + c0] = acc0[r] + bias0;
    xg[(size_t)grow * G4 + c1] = acc1[r] + bias1;
  }
}

// ---------------------------------------------------------------------------
// Kernel 2 (launched once per timestep): fused gates GEMM + LSTM cell.
//   gates[b, g] = xg_t[b, g] + h . W_hh[g, :]   (M=64, K=512)
//   Each WG owns 32 hidden units and gathers all 4 gate column groups
//   (columns j, j+H, j+2H, j+3H) so the cell fuses in LDS.
// ---------------------------------------------------------------------------
__global__ __launch_bounds__(256) void lstm_step(
    const float* __restrict__ xg_t,   // [B, 4H] for this t
    const float* __restrict__ W_hh,   // [4H, H]
    const float* __restrict__ hin, const float* __restrict__ cin,
    float* __restrict__ hout, float* __restrict__ cout,
    float* __restrict__ hsum,
    const float* __restrict__ mh_t, const float* __restrict__ mc_t)
{
  __shared__ float As[2][64][36];    // h tile [b][k]
  __shared__ float Bs[2][128][36];   // gathered W_hh rows [c][k], c = gate*32 + jj
  __shared__ float Gt[64][132];      // gate pre-activations [b][c]

  const int tid  = threadIdx.x;
  const int wv   = tid >> 5;
  const int lane = tid & 31;
  const int lrow = lane & 15;
  const int lhi  = lane >> 4;

  const int j0   = blockIdx.x * 32;                  // hidden-unit slice
  const int gate = wv & 3;
  const int half = wv >> 2;
  const int ccol = gate * 32 + half * 16 + lrow;     // gathered column 0..127
  const int gcol = gate * HH + j0 + half * 16 + lrow;

  auto stage = [&](int buf, int k0) {
#pragma unroll
    for (int i = 0; i < 2; ++i) {                    // h tile: 512 float4
      const int fi = tid + i * 256;
      const int r  = fi >> 3;
      const int kq = (fi & 7) * 4;
      async_b128(lds_off32(&As[buf][r][kq]), hin + (size_t)r * HH + k0 + kq);
    }
#pragma unroll
    for (int i = 0; i < 4; ++i) {                    // W_hh gather: 1024 float4
      const int fi = tid + i * 256;
      const int n  = fi >> 3;                        // 0..127
      const int kq = (fi & 7) * 4;
      const int col = (n >> 5) * HH + j0 + (n & 31); // global gate column
      async_b128(lds_off32(&Bs[buf][n][kq]), W_hh + (size_t)col * HH + k0 + kq);
    }
  };

  v8f acc[4];
#pragma unroll
  for (int mt = 0; mt < 4; ++mt)
#pragma unroll
    for (int r = 0; r < 8; ++r)
      acc[mt][r] = xg_t[(size_t)(mt * 16 + r + lhi * 8) * G4 + gcol];

  auto compute = [&](int buf) {
#pragma unroll
    for (int ks = 0; ks < 8; ++ks) {
      const int kk = ks * 4 + lhi * 2;
      const v2f b = *(const v2f*)&Bs[buf][ccol][kk];
#pragma unroll
      for (int mt = 0; mt < 4; ++mt) {
        const v2f a = *(const v2f*)&As[buf][mt * 16 + lrow][kk];
        acc[mt] = __builtin_amdgcn_wmma_f32_16x16x4_f32(false, a, false, b, (short)0, acc[mt], false, false);
      }
    }
  };

  stage(0, 0);
  for (int k0 = 0; k0 < HH - 32; k0 += 32) {
    const int buf = (k0 >> 5) & 1;
    stage(buf ^ 1, k0 + 32);
    WAITA(6);                 // 6 outstanding = next buffer; current buffer resident
    __syncthreads();
    compute(buf);
    __syncthreads();
  }
  WAITA(0);
  __syncthreads();
  compute(((HH - 32) >> 5) & 1);

#pragma unroll
  for (int mt = 0; mt < 4; ++mt)
#pragma unroll
    for (int r = 0; r < 8; ++r)
      Gt[mt * 16 + r + lhi * 8][ccol] = acc[mt][r];
  __syncthreads();

  // fused LSTM cell + dropout + running h-sum (each (b, j) owned by exactly one thread)
#pragma unroll
  for (int i = 0; i < 8; ++i) {
    const int idx = tid + i * 256;     // 0..2047 = 64 batch x 32 units
    const int b   = idx >> 5;
    const int jj  = idx & 31;
    const int j   = j0 + jj;
    const float ig = Gt[b][      jj];
    const float fg = Gt[b][32  + jj];
    const float gg = Gt[b][64  + jj];
    const float og = Gt[b][96  + jj];
    const float si = 1.0f / (1.0f + __expf(-ig));
    const float sf = 1.0f / (1.0f + __expf(-fg));
    const float so = 1.0f / (1.0f + __expf(-og));
    float cn = sf * cin[(size_t)b * HH + j] + si * tanhf(gg);
    float hn = so * tanhf(cn);
    hn *= mh_t[(size_t)b * HH + j];
    cn *= mc_t[(size_t)b * HH + j];
    hout[(size_t)b * HH + j]  = hn;
    cout[(size_t)b * HH + j]  = cn;
    hsum[(size_t)b * HH + j] += hn;
  }
}

// ---------------------------------------------------------------------------
// Init + head. mean_t(h_t @ W_out^T + b) == (sum_t h_t / T) @ W_out^T + b.
// ---------------------------------------------------------------------------
__global__ void init_state(const float* __restrict__ h0, const float* __restrict__ c0,
                           float* __restrict__ h, float* __restrict__ c,
                           float* __restrict__ hs)
{
  const int i = blockIdx.x * blockDim.x + threadIdx.x;
  if (i < BB * HH) { h[i] = h0[i]; c[i] = c0[i]; hs[i] = 0.0f; }
}

__global__ void head_gemm(const float* __restrict__ hs, const float* __restrict__ W_out,
                          const float* __restrict__ b_out, float* __restrict__ out)
{
  const int idx = blockIdx.x * blockDim.x + threadIdx.x;
  if (idx >= BB * NC) return;
  const int b = idx / NC, n = idx % NC;
  float s = 0.0f;
  for (int j = 0; j < HH; ++j) s += hs[b * HH + j] * W_out[n * HH + j];
  out[idx] = s * (1.0f / TT) + b_out[n];
}

// ---------------------------------------------------------------------------
extern "C" void kernel_launch(void* const* d_in, const int* in_sizes, int n_in,
                              void* d_out, int out_size, void* d_ws, size_t ws_size,
                              hipStream_t stream) {
  const float* x      = (const float*)d_in[0];
  const float* h0     = (const float*)d_in[1];
  const float* c0     = (const float*)d_in[2];
  const float* W_ih   = (const float*)d_in[3];
  const float* W_hh   = (const float*)d_in[4];
  const float* b_ih   = (const float*)d_in[5];
  const float* b_hh   = (const float*)d_in[6];
  const float* W_out  = (const float*)d_in[7];
  const float* b_out  = (const float*)d_in[8];
  const float* mask_h = (const float*)d_in[9];
  const float* mask_c = (const float*)d_in[10];

  float* ws   = (float*)d_ws;
  float* xg   = ws;                                  // T*B*4H = 16,777,216 floats (64 MB)
  float* hbuf = xg + (size_t)TT * BB * G4;           // 2 * B*H (double buffer)
  float* cbuf = hbuf + 2 * BB * HH;                  // 2 * B*H
  float* hsum = cbuf + 2 * BB * HH;                  // B*H

  init_state<<<(BB * HH + 255) / 256, 256, 0, stream>>>(h0, c0, hbuf, cbuf, hsum);

  dim3 g1(G4 / 64, (TT * BB) / 64);                  // 32 x 128 workgroups
  xgates_gemm<<<g1, 256, 0, stream>>>(x, W_ih, b_ih, b_hh, xg);

  for (int t = 0; t < TT; ++t) {
    const float* hin  = hbuf + (t & 1) * BB * HH;
    float*       hout = hbuf + ((t + 1) & 1) * BB * HH;
    const float* cin  = cbuf + (t & 1) * BB * HH;
    float*       cout = cbuf + ((t + 1) & 1) * BB * HH;
    lstm_step<<<HH / 32, 256, 0, stream>>>(
        xg + (size_t)t * BB * G4, W_hh, hin, cin, hout, cout, hsum,
        mask_h + (size_t)t * BB * HH, mask_c + (size_t)t * BB * HH);
  }

  head_gemm<<<(BB * NC + 255) / 256, 256, 0, stream>>>(hsum, W_out, b_out, (float*)d_out);
}